// CostVolume_11012296147552
// MI455X (gfx1250) — compile-verified
//
#include <hip/hip_runtime.h>
#include <stdint.h>

typedef uint32_t u32;
typedef uint64_t u64;
typedef u32   u32x4 __attribute__((ext_vector_type(4)));
typedef u32   u32x8 __attribute__((ext_vector_type(8)));
typedef float f32x4 __attribute__((ext_vector_type(4)));

#define B_   2
#define C_   32
#define H_   128
#define W_   240
#define D_   48
#define W4_  60                       // W/4 float4 columns per row
#define ROWSTRIDE (H_ * W_)           // 30720: out stride between consecutive d
#define CHSTRIDE  ((u64)D_ * ROWSTRIDE)
#define HALF_OFF  ((u64)32 * CHSTRIDE)  // offset from left half to right half of channel dim

// ---------------------------------------------------------------------------
// Tensor Data Mover: DMA one row of `nelem` f32 from global memory into LDS.
// Builds a 2-D D# (groups 0 and 1; groups 2/3 omitted -> NULL) per CDNA5 ISA
// ch.8 and issues TENSOR_LOAD_TO_LDS via inline asm (portable across both
// toolchains). Tracked by TENSORcnt.
// ---------------------------------------------------------------------------
__device__ __forceinline__ void tdm_load_row_f32(const float* gsrc, u32 lds_byte_off, u32 nelem)
{
    const u64 ga = (u64)(uintptr_t)gsrc;

    u32x4 g0;
    g0[0] = 1u;                                                // count=1, user mode, no gather
    g0[1] = lds_byte_off;                                      // lds_addr (bytes)
    g0[2] = (u32)(ga & 0xFFFFFFFFull);                         // global_addr[31:0]
    g0[3] = (u32)((ga >> 32) & 0x01FFFFFFull) | 0x80000000u;   // global_addr[56:32] | type=2

    u32x8 g1;
    g1[0] = 2u << 16;                                // workgroup_mask=0 (not in cluster), data_size=4B
    g1[1] = (nelem & 0xFFFFu) << 16;                 // tensor_dim0[15:0]   (bits 63:48)
    g1[2] = ((nelem >> 16) & 0xFFFFu) | (1u << 16);  // tensor_dim0[31:16], tensor_dim1=1
    g1[3] = (nelem & 0xFFFFu) << 16;                 // tensor_dim1[31:16]=0, tile_dim0=nelem
    g1[4] = 1u;                                      // tile_dim1=1, tile_dim2=0
    g1[5] = nelem;                                   // tensor_dim0_stride[31:0]
    g1[6] = 0u;                                      // stride0[47:32]=0, tensor_dim1_stride lo=0
    g1[7] = 0u;                                      // tensor_dim1_stride hi=0

    asm volatile("tensor_load_to_lds %0, %1" :: "s"(g0), "s"(g1) : "memory");
}

__global__ void __launch_bounds__(256)
costvol_kernel(const float* __restrict__ left,
               const float* __restrict__ right,
               float* __restrict__ out)
{
    __shared__ __align__(16) float sL[W_];
    __shared__ __align__(16) float sR[W_];

    const int tid = threadIdx.x;
    const int bi  = blockIdx.x;          // one block per (b, c, h) row
    const int h   = bi & (H_ - 1);
    const int t   = bi >> 7;
    const int c   = t & (C_ - 1);
    const int b   = t >> 5;

    const u64 inRow = ((u64)(b * C_ + c) * H_ + h) * W_;

    // Wave 0 stages both input rows into LDS via the TDM, then waits TENSORcnt.
    if (tid < 32) {
        tdm_load_row_f32(left  + inRow, (u32)(uintptr_t)&sL[0], W_);
        tdm_load_row_f32(right + inRow, (u32)(uintptr_t)&sR[0], W_);
        __builtin_amdgcn_s_wait_tensorcnt(0);
    }
    __syncthreads();

    // out[((b*64 + ch)*D + d)*H + h]*W + w ; ch = c (+32 for right half)
    const u64 baseL = (u64)(b * 64 + c) * CHSTRIDE + (u64)h * W_;

    // (d, half, w4) with w4 padded to 64 so each 32-lane wave holds uniform
    // (d, half) and stores one contiguous run of float4s.
    for (int u = tid; u < D_ * 2 * 64; u += 256) {
        const int w4 = u & 63;
        if (w4 >= W4_) continue;
        const int r    = u >> 6;
        const int half = r & 1;          // 0 = left volume, 1 = right volume
        const int d    = r >> 1;
        const int w0   = w4 << 2;

        f32x4 v;
        if (half == 0) {
            const f32x4 s = *(const f32x4*)&sL[w0];   // aligned ds_load_b128
            v.x = (w0     >= d) ? s.x : 0.0f;
            v.y = (w0 + 1 >= d) ? s.y : 0.0f;
            v.z = (w0 + 2 >= d) ? s.z : 0.0f;
            v.w = (w0 + 3 >= d) ? s.w : 0.0f;
        } else {
            int i0 = w0     - d; i0 = i0 < 0 ? 0 : i0;
            int i1 = w0 + 1 - d; i1 = i1 < 0 ? 0 : i1;
            int i2 = w0 + 2 - d; i2 = i2 < 0 ? 0 : i2;
            int i3 = w0 + 3 - d; i3 = i3 < 0 ? 0 : i3;
            v.x = (w0     >= d) ? sR[i0] : 0.0f;
            v.y = (w0 + 1 >= d) ? sR[i1] : 0.0f;
            v.z = (w0 + 2 >= d) ? sR[i2] : 0.0f;
            v.w = (w0 + 3 >= d) ? sR[i3] : 0.0f;
        }

        const u64 off = baseL + (half ? HALF_OFF : (u64)0) + (u64)d * ROWSTRIDE + (u64)w0;
        __builtin_nontemporal_store(v, (f32x4*)(out + off));  // streaming 755MB: NT store
    }
}

extern "C" void kernel_launch(void* const* d_in, const int* in_sizes, int n_in,
                              void* d_out, int out_size, void* d_ws, size_t ws_size,
                              hipStream_t stream)
{
    const float* left  = (const float*)d_in[0];
    const float* right = (const float*)d_in[1];
    // d_in[2] = max_disp (192) -> D = 48, baked into the kernel constants.
    float* out = (float*)d_out;

    dim3 grid(B_ * C_ * H_);   // 8192 blocks: one per (b, c, h) input row
    dim3 block(256);
    costvol_kernel<<<grid, block, 0, stream>>>(left, right, out);
}